// Eggtart_54271206752365
// MI455X (gfx1250) — compile-verified
//
#include <hip/hip_runtime.h>
#include <hip/hip_bf16.h>

typedef __attribute__((ext_vector_type(2))) float v2f;
typedef __attribute__((ext_vector_type(8))) float v8f;

#define RES      64
#define NPTS     1024
#define NBATCH   32
#define WAVES_PER_BLK 8

// -1/(2*sigma^2) * log2(e) = -5000 * 1.4426950408889634
#define SCALE_LOG2E (-7213.475204444817f)

__global__ __launch_bounds__(256) void
gauss_raster_wmma(const float* __restrict__ x, float* __restrict__ out)
{
    const int lane   = threadIdx.x & 31;
    const int wid    = blockIdx.x * WAVES_PER_BLK + (threadIdx.x >> 5); // 0..8191
    const int batch  = wid >> 8;          // 0..31
    const int tile   = wid & 255;         // 0..255 (16-pixel tiles, 4 per grid row)
    const int row_i  = tile >> 2;         // 0..63
    const int col0   = (tile & 3) << 4;   // 0,16,32,48
    const int nIdx   = lane & 15;         // pixel-within-tile / point-within-tile

    // Meshgrid regenerated per the reference (double math, cast to f32):
    //   xm_j = (-16 + 2.5 j)/64 ;  ym_i = (-25.6 + 2.2*(63-i))/64
    const float xm = (float)((-16.0 + 2.5 * (double)(col0 + nIdx)) / 64.0);
    const float ym = (float)((-25.6 + 2.2 * (double)(63 - row_i)) / 64.0);

    // B fragment (K=4 x N=16, loop-invariant). 2 VGPRs:
    //   v0: lanes0-15 -> K0 = -2*xm   ; lanes16-31 -> K2 = 1.0
    //   v1: lanes0-15 -> K1 = -2*ym   ; lanes16-31 -> K3 = xm^2+ym^2
    v2f bfrag;
    if (lane < 16) {
        bfrag.x = -2.0f * xm;
        bfrag.y = -2.0f * ym;
    } else {
        bfrag.x = 1.0f;
        bfrag.y = xm * xm + ym * ym;
    }

    const float* xb = x + batch * (NPTS * 2);

    v8f acc = {0.f, 0.f, 0.f, 0.f, 0.f, 0.f, 0.f, 0.f};
    const v8f zero = {0.f, 0.f, 0.f, 0.f, 0.f, 0.f, 0.f, 0.f};

    for (int pt = 0; pt < NPTS; pt += 16) {
        // Each lane fetches point (pt + nIdx); lane-halves hit the same lines.
        const float2 p = *(const float2*)(xb + (pt + nIdx) * 2);

        // A fragment (M=16 points x K=4), scale s = -5000*log2(e) folded in:
        //   lanes0-15:  v0 = s*x, v1 = s*y       (K0,K1 for M=lane)
        //   lanes16-31: v0 = s*(x^2+y^2), v1 = s (K2,K3 for M=lane-16)
        v2f afrag;
        if (lane < 16) {
            afrag.x = SCALE_LOG2E * p.x;
            afrag.y = SCALE_LOG2E * p.y;
        } else {
            afrag.x = SCALE_LOG2E * (p.x * p.x + p.y * p.y);
            afrag.y = SCALE_LOG2E;
        }

        // D[m,n] = s * dist2(point m, pixel n)  == log2 of the gaussian
        v8f d = __builtin_amdgcn_wmma_f32_16x16x4_f32(
            /*neg_a=*/false, afrag, /*neg_b=*/false, bfrag,
            /*c_mod=*/(short)0, zero, /*reuse_a=*/false, /*reuse_b=*/false);

        // One v_exp_f32 + one v_add_f32 per pair; co-executes with next WMMA.
        #pragma unroll
        for (int r = 0; r < 8; ++r)
            acc[r] += __builtin_amdgcn_exp2f(d[r]);
    }

    // Per-lane: acc[r] covers point M=r (lanes 0-15) / M=r+8 (lanes 16-31),
    // pixel N = lane&15. Reduce 8 VGPRs then fold the two lane halves.
    float tot = ((acc[0] + acc[1]) + (acc[2] + acc[3])) +
                ((acc[4] + acc[5]) + (acc[6] + acc[7]));
    tot += __shfl_xor(tot, 16, 32);

    if (lane < 16)
        out[batch * (RES * RES) + row_i * RES + col0 + lane] = tot;
}

extern "C" void kernel_launch(void* const* d_in, const int* in_sizes, int n_in,
                              void* d_out, int out_size, void* d_ws, size_t ws_size,
                              hipStream_t stream)
{
    (void)in_sizes; (void)n_in; (void)d_ws; (void)ws_size; (void)out_size;
    const float* x  = (const float*)d_in[0];
    float* out      = (float*)d_out;

    const int total_waves = NBATCH * 256;              // 8192
    const int blocks = total_waves / WAVES_PER_BLK;    // 1024
    gauss_raster_wmma<<<blocks, WAVES_PER_BLK * 32, 0, stream>>>(x, out);
}